// GATResBlock_27625229648502
// MI455X (gfx1250) — compile-verified
//
#include <hip/hip_runtime.h>

// ---------------------------------------------------------------------------
// GAT residual block for MI455X (gfx1250, wave32).
//   layer: h = in @ W (WMMA f32) ; a_src/a_dst dots ; edge softmax (3 passes) ;
//          aggregate ; batchnorm (+ELU or +ReLU/residual)
// ---------------------------------------------------------------------------

#define DCH 256   // channels

typedef __attribute__((ext_vector_type(2))) float v2f;
typedef __attribute__((ext_vector_type(8))) float v8f;

// ------------------------- helpers -----------------------------------------

__device__ __forceinline__ unsigned enc_f32(float f) {
    unsigned u = __float_as_uint(f);
    return (u & 0x80000000u) ? ~u : (u | 0x80000000u);   // monotone float->uint
}
__device__ __forceinline__ float dec_f32(unsigned u) {
    return (u & 0x80000000u) ? __uint_as_float(u & 0x7FFFFFFFu)
                             : __uint_as_float(~u);
}
__device__ __forceinline__ void edge_sd(int e, const int* __restrict__ ei,
                                        int E, int& s, int& d) {
    if (e < E) { s = ei[e]; d = ei[E + e]; }
    else       { s = e - E; d = e - E; }                  // self loops
}

// ------------------------- zero scratch ------------------------------------

__global__ void __launch_bounds__(256) zero_f32(float* __restrict__ p, int n) {
    int i = blockIdx.x * 256 + threadIdx.x;
    if (i < n) p[i] = 0.0f;
}

// ------------------------- GEMM: H = X @ W  (fp32 WMMA) --------------------
// Block = 128 threads = 4 waves. Wave w computes rows [16*bx,16*bx+16) x
// cols [64*w, 64*w+64) as four 16x16 WMMA tiles, K stepped by 4.

__global__ void __launch_bounds__(128)
gemm_wmma_f32(const float* __restrict__ X, const float* __restrict__ W,
              float* __restrict__ H) {
    const int lane  = threadIdx.x & 31;
    const int wave  = threadIdx.x >> 5;
    const int m0    = blockIdx.x << 4;
    const int nbase = wave << 6;                 // 0,64,128,192
    const int mrow  = lane & 15;
    const int khalf = (lane >> 4) << 1;          // 0 or 2 (ISA A-frag layout)

    const float* __restrict__ Arow = X + (size_t)(m0 + mrow) * DCH;
    const float* __restrict__ Bc   = W + nbase + (lane & 15);

    v8f acc0 = {}, acc1 = {}, acc2 = {}, acc3 = {};

    #pragma unroll 4
    for (int k = 0; k < DCH; k += 4) {
        v2f a = *(const v2f*)(Arow + k + khalf);          // A[m][k+khalf..+1]
        const float* bp = Bc + (size_t)(k + khalf) * DCH; // B rows k+khalf,+1
        v2f b0, b1, b2, b3;
        b0.x = bp[0];   b0.y = bp[DCH + 0];
        b1.x = bp[16];  b1.y = bp[DCH + 16];
        b2.x = bp[32];  b2.y = bp[DCH + 32];
        b3.x = bp[48];  b3.y = bp[DCH + 48];
        acc0 = __builtin_amdgcn_wmma_f32_16x16x4_f32(false, a, false, b0,
                                                     (short)0, acc0, false, false);
        acc1 = __builtin_amdgcn_wmma_f32_16x16x4_f32(false, a, false, b1,
                                                     (short)0, acc1, false, false);
        acc2 = __builtin_amdgcn_wmma_f32_16x16x4_f32(false, a, false, b2,
                                                     (short)0, acc2, false, false);
        acc3 = __builtin_amdgcn_wmma_f32_16x16x4_f32(false, a, false, b3,
                                                     (short)0, acc3, false, false);
    }

    // C/D layout: VGPR r -> M = r + (lane>=16 ? 8 : 0), N = lane&15
    const int mtop = m0 + ((lane >> 4) << 3);
    float* __restrict__ Hp = H + (size_t)mtop * DCH + nbase + (lane & 15);
    #pragma unroll
    for (int r = 0; r < 8; ++r) {
        float* row = Hp + (size_t)r * DCH;
        row[0]  = acc0[r];
        row[16] = acc1[r];
        row[32] = acc2[r];
        row[48] = acc3[r];
    }
}

// ------------------------- attention scores --------------------------------
// One wave per node: a_src[n] = dot(h[n], att_src), a_dst likewise.

__global__ void __launch_bounds__(256)
attn_scores(const float* __restrict__ H, const float* __restrict__ att_s,
            const float* __restrict__ att_d, float* __restrict__ a_src,
            float* __restrict__ a_dst, int Nn) {
    const int lane = threadIdx.x & 31;
    const int wave = threadIdx.x >> 5;
    const int n = blockIdx.x * 8 + wave;
    if (n >= Nn) return;
    const float4* hp = (const float4*)(H + (size_t)n * DCH);
    const float4* sp = (const float4*)att_s;
    const float4* dp = (const float4*)att_d;
    float ss = 0.f, sd = 0.f;
    #pragma unroll
    for (int i = 0; i < 2; ++i) {
        int idx = lane + 32 * i;                 // 64 float4 = 256 floats
        float4 h4 = hp[idx], s4 = sp[idx], d4 = dp[idx];
        ss += h4.x * s4.x + h4.y * s4.y + h4.z * s4.z + h4.w * s4.w;
        sd += h4.x * d4.x + h4.y * d4.y + h4.z * d4.z + h4.w * d4.w;
    }
    #pragma unroll
    for (int off = 16; off > 0; off >>= 1) {
        ss += __shfl_xor(ss, off, 32);
        sd += __shfl_xor(sd, off, 32);
    }
    if (lane == 0) { a_src[n] = ss; a_dst[n] = sd; }
}

// ------------------------- edge softmax pass 1: segment max ----------------

__global__ void __launch_bounds__(256)
edge_max(const int* __restrict__ ei, int E, int ET,
         const float* __restrict__ a_src, const float* __restrict__ a_dst,
         unsigned* __restrict__ emax) {
    int e = blockIdx.x * 256 + threadIdx.x;
    if (e >= ET) return;
    int s, d; edge_sd(e, ei, E, s, d);
    float v = a_src[s] + a_dst[d];
    v = (v > 0.f) ? v : 0.2f * v;                // leaky_relu(0.2)
    atomicMax(&emax[d], enc_f32(v));
}

// ------------------------- edge softmax pass 2: exp + denom ----------------

__global__ void __launch_bounds__(256)
edge_exp(const int* __restrict__ ei, int E, int ET,
         const float* __restrict__ a_src, const float* __restrict__ a_dst,
         const unsigned* __restrict__ emax, float* __restrict__ exbuf,
         float* __restrict__ denom) {
    int e = blockIdx.x * 256 + threadIdx.x;
    if (e >= ET) return;
    int s, d; edge_sd(e, ei, E, s, d);
    float v = a_src[s] + a_dst[d];
    v = (v > 0.f) ? v : 0.2f * v;
    float ex = __expf(v - dec_f32(emax[d]));
    exbuf[e] = ex;
    atomicAdd(&denom[d], ex);
}

// ------------------------- edge softmax pass 3: aggregate ------------------
// One wave per edge; each lane handles 8 coalesced channels.

__global__ void __launch_bounds__(256)
edge_aggregate(const int* __restrict__ ei, int E, int ET,
               const float* __restrict__ H, const float* __restrict__ exbuf,
               const float* __restrict__ denom, float* __restrict__ out) {
    const int lane = threadIdx.x & 31;
    const int wave = threadIdx.x >> 5;
    int e = blockIdx.x * 8 + wave;
    if (e >= ET) return;
    int s, d; edge_sd(e, ei, E, s, d);
    float alpha = exbuf[e] / denom[d];
    const float* __restrict__ hs = H + (size_t)s * DCH;
    float* __restrict__ od = out + (size_t)d * DCH;
    #pragma unroll
    for (int i = 0; i < 8; ++i) {
        int c = lane + 32 * i;
        atomicAdd(&od[c], hs[c] * alpha);
    }
}

// ------------------------- batchnorm stats ---------------------------------
// 256 threads; thread t owns channel t, accumulates over a row chunk.

__global__ void __launch_bounds__(256)
bn_stats(const float* __restrict__ X, int Nn, int rowsPerBlock,
         float* __restrict__ gsum, float* __restrict__ gsumsq) {
    const int c  = threadIdx.x;
    int r0 = blockIdx.x * rowsPerBlock;
    int r1 = r0 + rowsPerBlock; if (r1 > Nn) r1 = Nn;
    float s = 0.f, s2 = 0.f;
    for (int r = r0; r < r1; ++r) {
        float v = X[(size_t)r * DCH + c];
        s += v; s2 += v * v;
    }
    atomicAdd(&gsum[c], s);
    atomicAdd(&gsumsq[c], s2);
}

// ------------------------- batchnorm apply + ELU (layer 1) -----------------

__global__ void __launch_bounds__(256)
bn_elu(const float* __restrict__ X, const float* __restrict__ gsum,
       const float* __restrict__ gsumsq, const float* __restrict__ gamma,
       const float* __restrict__ beta, float* __restrict__ Y,
       int total, float invN) {
    int i = blockIdx.x * 256 + threadIdx.x;
    if (i >= total) return;
    int c = i & (DCH - 1);
    float m   = gsum[c] * invN;
    float var = fmaf(-m, m, gsumsq[c] * invN);
    float rs  = rsqrtf(var + 1e-5f);
    float y   = (X[i] - m) * rs * gamma[c] + beta[c];
    Y[i] = (y > 0.f) ? y : (__expf(y) - 1.f);            // ELU(alpha=1)
}

// --------------- batchnorm apply + ReLU + residual + ReLU (layer 2) --------

__global__ void __launch_bounds__(256)
bn_relu_res(const float* __restrict__ X, const float* __restrict__ gsum,
            const float* __restrict__ gsumsq, const float* __restrict__ gamma,
            const float* __restrict__ beta, const float* __restrict__ resid,
            float* __restrict__ out, int total, float invN) {
    int i = blockIdx.x * 256 + threadIdx.x;
    if (i >= total) return;
    int c = i & (DCH - 1);
    float m   = gsum[c] * invN;
    float var = fmaf(-m, m, gsumsq[c] * invN);
    float rs  = rsqrtf(var + 1e-5f);
    float y   = (X[i] - m) * rs * gamma[c] + beta[c];
    y = fmaxf(y, 0.f);
    out[i] = fmaxf(y + resid[i], 0.f);
}

// ---------------------------------------------------------------------------

extern "C" void kernel_launch(void* const* d_in, const int* in_sizes, int n_in,
                              void* d_out, int out_size, void* d_ws, size_t ws_size,
                              hipStream_t stream) {
    const int Nn = in_sizes[0] / DCH;   // 10000
    const int E  = in_sizes[1] / 2;     // 320000
    const int ET = E + Nn;

    const float* x        = (const float*)d_in[0];
    const int*   ei       = (const int*)d_in[1];
    const float* W1       = (const float*)d_in[2];
    const float* att_src1 = (const float*)d_in[3];
    const float* att_dst1 = (const float*)d_in[4];
    const float* gamma1   = (const float*)d_in[6];
    const float* beta1    = (const float*)d_in[7];
    const float* W2       = (const float*)d_in[8];
    const float* att_src2 = (const float*)d_in[9];
    const float* att_dst2 = (const float*)d_in[10];
    const float* gamma2   = (const float*)d_in[12];
    const float* beta2    = (const float*)d_in[13];
    // bias1/bias2 (d_in[5], d_in[11]) cancel exactly inside batchnorm — skipped.
    float* out = (float*)d_out;

    // ---- workspace layout (bytes, 256-aligned chunks) ----
    char* ws = (char*)d_ws;
    size_t off = 0;
    auto carve = [&](size_t bytes) -> char* {
        char* p = ws + off;
        off += (bytes + 255) & ~(size_t)255;
        return p;
    };
    float*    hbuf   = (float*)carve((size_t)Nn * DCH * 4);
    float*    actbuf = (float*)carve((size_t)Nn * DCH * 4);
    float*    exbuf  = (float*)carve((size_t)ET * 4);
    float*    a_src  = (float*)carve((size_t)Nn * 4);
    float*    a_dst  = (float*)carve((size_t)Nn * 4);
    // contiguous zero-block: aggbuf | denom | emax | gsum | gsumsq
    size_t zeroFloats = (size_t)Nn * DCH + Nn + Nn + DCH + DCH;
    float*    zblock = (float*)carve(zeroFloats * 4);
    float*    aggbuf = zblock;
    float*    denom  = aggbuf + (size_t)Nn * DCH;
    unsigned* emax   = (unsigned*)(denom + Nn);
    float*    gsum   = (float*)(emax + Nn);
    float*    gsumsq = gsum + DCH;

    const int total   = Nn * DCH;
    const float invN  = 1.0f / (float)Nn;
    const int rowsPB  = 64;

    dim3 blk256(256);
    const int gZero  = (int)((zeroFloats + 255) / 256);
    const int gGemm  = Nn / 16;                 // 625 (exact)
    const int gAttn  = (Nn + 7) / 8;
    const int gEdgeT = (ET + 255) / 256;
    const int gEdgeW = (ET + 7) / 8;
    const int gStats = (Nn + rowsPB - 1) / rowsPB;
    const int gElem  = (total + 255) / 256;

    // =================== layer 1 ===================
    zero_f32<<<gZero, blk256, 0, stream>>>(zblock, (int)zeroFloats);
    gemm_wmma_f32<<<gGemm, dim3(128), 0, stream>>>(x, W1, hbuf);
    attn_scores<<<gAttn, blk256, 0, stream>>>(hbuf, att_src1, att_dst1,
                                              a_src, a_dst, Nn);
    edge_max<<<gEdgeT, blk256, 0, stream>>>(ei, E, ET, a_src, a_dst, emax);
    edge_exp<<<gEdgeT, blk256, 0, stream>>>(ei, E, ET, a_src, a_dst, emax,
                                            exbuf, denom);
    edge_aggregate<<<gEdgeW, blk256, 0, stream>>>(ei, E, ET, hbuf, exbuf,
                                                  denom, aggbuf);
    bn_stats<<<gStats, blk256, 0, stream>>>(aggbuf, Nn, rowsPB, gsum, gsumsq);
    bn_elu<<<gElem, blk256, 0, stream>>>(aggbuf, gsum, gsumsq, gamma1, beta1,
                                         actbuf, total, invN);

    // =================== layer 2 ===================
    zero_f32<<<gZero, blk256, 0, stream>>>(zblock, (int)zeroFloats);
    gemm_wmma_f32<<<gGemm, dim3(128), 0, stream>>>(actbuf, W2, hbuf);
    attn_scores<<<gAttn, blk256, 0, stream>>>(hbuf, att_src2, att_dst2,
                                              a_src, a_dst, Nn);
    edge_max<<<gEdgeT, blk256, 0, stream>>>(ei, E, ET, a_src, a_dst, emax);
    edge_exp<<<gEdgeT, blk256, 0, stream>>>(ei, E, ET, a_src, a_dst, emax,
                                            exbuf, denom);
    edge_aggregate<<<gEdgeW, blk256, 0, stream>>>(ei, E, ET, hbuf, exbuf,
                                                  denom, aggbuf);
    bn_stats<<<gStats, blk256, 0, stream>>>(aggbuf, Nn, rowsPB, gsum, gsumsq);
    bn_relu_res<<<gElem, blk256, 0, stream>>>(aggbuf, gsum, gsumsq, gamma2,
                                              beta2, x, out, total, invN);
}